// GraphConvolution_44066364456987
// MI455X (gfx1250) — compile-verified
//
#include <hip/hip_runtime.h>
#include <hip/hip_bf16.h>
#include <stdint.h>

typedef __attribute__((ext_vector_type(2))) float v2f;
typedef __attribute__((ext_vector_type(8))) float v8f;

#define D 256            // D_IN == D_OUT == 256
#define LDS_STRIDE 260   // 260 % 64 == 4 -> 16 lanes hit distinct banks; 1040B row pitch keeps 16B alignment

// -------------------------------------------------------------------------
// Kernel 1: support = X @ W with V_WMMA_F32_16X16X4_F32 (full f32 precision).
// Block = 256 threads = 8 waves; tile 16 rows x 128 cols; wave w owns the
// 16x16 tile at col blockIdx.y*128 + w*16. The 16x256 A tile is staged in
// LDS with CDNA5 async global->LDS loads (ASYNCcnt), shared by all waves.
// W (256KB) is WGP$/L2 resident; B fragments load directly from global.
// -------------------------------------------------------------------------
__global__ __launch_bounds__(256) void gcn_gemm_wmma_f32(
    const float* __restrict__ X, const float* __restrict__ W,
    float* __restrict__ S, int N)
{
    __shared__ float As[16 * LDS_STRIDE];
    const int tid  = threadIdx.x;
    const int lane = tid & 31;
    const int wave = tid >> 5;
    const int tile_m = blockIdx.x * 16;
    const int tile_n = blockIdx.y * 128 + wave * 16;

    // Async-stage the 16x256 A tile (1024 float4; 4 per thread). Rows past N
    // are clamped (their results are never stored).
    #pragma unroll
    for (int i = 0; i < 4; ++i) {
        int f  = tid + i * 256;   // float4 index within tile
        int r  = f >> 6;          // 64 float4 per row
        int cv = f & 63;
        int gr = tile_m + r; if (gr >= N) gr = N - 1;
        const float* gptr = X + (size_t)gr * D + cv * 4;
        unsigned ldsoff = (unsigned)(uintptr_t)&As[r * LDS_STRIDE + cv * 4]; // 16B aligned
        asm volatile("global_load_async_to_lds_b128 %0, %1, off"
                     :: "v"(ldsoff), "v"((uint64_t)(uintptr_t)gptr) : "memory");
    }
    asm volatile("s_wait_asynccnt 0x0" ::: "memory");
    __syncthreads();

    // ISA 32-bit A 16x4 layout: lanes 0-15 -> K={k,k+1}, lanes 16-31 -> K={k+2,k+3}
    const int m     = lane & 15;
    const int khalf = (lane >> 4) * 2;
    const int n     = tile_n + (lane & 15);

    v8f c = {};
    #pragma unroll 16
    for (int k = 0; k < D; k += 4) {
        v2f a = *(const v2f*)&As[m * LDS_STRIDE + k + khalf];   // 8B-aligned ds_load
        v2f b;
        b.x = W[(size_t)(k + khalf)     * D + n];
        b.y = W[(size_t)(k + khalf + 1) * D + n];
        c = __builtin_amdgcn_wmma_f32_16x16x4_f32(
                false, a, false, b, (short)0, c, false, false);
    }

    // 16x16 f32 C/D layout: VGPR r -> M=r (lanes 0-15) / M=r+8 (lanes 16-31)
    #pragma unroll
    for (int r = 0; r < 8; ++r) {
        int row = tile_m + r + (lane >> 4) * 8;
        if (row < N) S[(size_t)row * D + tile_n + (lane & 15)] = c[r];
    }
}

// ------------------------- CSR build pipeline ----------------------------
__global__ void gcn_zero_i32(int* __restrict__ p, int n)
{
    int i = blockIdx.x * blockDim.x + threadIdx.x;
    if (i < n) p[i] = 0;
}

__global__ void gcn_edge_histogram(const int* __restrict__ erow,
                                   int* __restrict__ counts, int E)
{
    int e = blockIdx.x * blockDim.x + threadIdx.x;
    if (e < E) atomicAdd(&counts[erow[e]], 1);
}

// Single-workgroup exclusive scan of counts[0..N) -> starts[0..N]; cursor copy.
__global__ __launch_bounds__(1024) void gcn_scan_starts(
    const int* __restrict__ counts, int* __restrict__ starts,
    int* __restrict__ cursor, int N)
{
    __shared__ int buf[1024];
    __shared__ int carry;
    const int tid = threadIdx.x;
    if (tid == 0) carry = 0;
    __syncthreads();
    for (int base = 0; base < N; base += 1024) {
        int i = base + tid;
        int x = (i < N) ? counts[i] : 0;
        buf[tid] = x;
        __syncthreads();
        #pragma unroll
        for (int off = 1; off < 1024; off <<= 1) {        // Hillis-Steele inclusive
            int t = (tid >= off) ? buf[tid - off] : 0;
            __syncthreads();
            buf[tid] += t;
            __syncthreads();
        }
        int incl = buf[tid];
        int c = carry;
        if (i < N) { starts[i] = c + incl - x; cursor[i] = c + incl - x; }
        __syncthreads();
        if (tid == 1023) carry = c + incl;
        __syncthreads();
    }
    if (tid == 0) starts[N] = carry;   // == E
}

__global__ void gcn_bin_edges(const int* __restrict__ erow,
                              const int* __restrict__ ecol,
                              const float* __restrict__ eval,
                              int* __restrict__ cursor,
                              int* __restrict__ bcol, float* __restrict__ bval, int E)
{
    int e = blockIdx.x * blockDim.x + threadIdx.x;
    if (e < E) {
        int pos = atomicAdd(&cursor[erow[e]], 1);
        bcol[pos] = ecol[e];
        bval[pos] = eval[e];
    }
}

// -------------------------------------------------------------------------
// Kernel 2: out[r] = bias + sum_{e in row r} val[e] * support[col[e]].
// One wave32 per row; 8 f32 accumulators per lane; zero atomics; all
// support gathers are L2-resident (100MB table in 192MB L2).
// -------------------------------------------------------------------------
__global__ __launch_bounds__(256) void gcn_row_gather(
    const float* __restrict__ support, const int* __restrict__ starts,
    const int* __restrict__ bcol, const float* __restrict__ bval,
    const float* __restrict__ bias, float* __restrict__ out, int N)
{
    const int lane = threadIdx.x & 31;
    const int row  = (blockIdx.x * blockDim.x + threadIdx.x) >> 5;
    if (row >= N) return;

    const int beg = starts[row], end = starts[row + 1];
    float4 acc0 = ((const float4*)bias)[lane];        // dims [lane*4 .. +3]
    float4 acc1 = ((const float4*)bias)[lane + 32];   // dims [128+lane*4 ..]

    for (int e = beg; e < end; ++e) {
        const int   col = bcol[e];
        const float v   = bval[e];
        if (e + 1 < end)   // hide the random-gather latency of the next row
            __builtin_prefetch(support + (size_t)bcol[e + 1] * D + (lane & 7) * 32, 0, 0);
        const float4* src = (const float4*)(support + (size_t)col * D);
        float4 a = src[lane];
        float4 b = src[lane + 32];
        acc0.x += a.x * v; acc0.y += a.y * v; acc0.z += a.z * v; acc0.w += a.w * v;
        acc1.x += b.x * v; acc1.y += b.y * v; acc1.z += b.z * v; acc1.w += b.w * v;
    }
    float4* dst = (float4*)(out + (size_t)row * D);
    dst[lane]      = acc0;
    dst[lane + 32] = acc1;
}

// ------------------- fallback path (workspace too small) -----------------
__global__ void gcn_init_bias(float4* __restrict__ out4,
                              const float4* __restrict__ bias4, int nVec4)
{
    int i = blockIdx.x * blockDim.x + threadIdx.x;
    if (i < nVec4) out4[i] = bias4[i & 63];
}

__global__ __launch_bounds__(256) void gcn_edge_scatter_atomic(
    const float* __restrict__ support, const int* __restrict__ erow,
    const int* __restrict__ ecol, const float* __restrict__ eval,
    float* __restrict__ out, int nEdges)
{
    const int lane = threadIdx.x & 31;
    const int wid  = (blockIdx.x * blockDim.x + threadIdx.x) >> 5;
    const int nw   = (gridDim.x * blockDim.x) >> 5;
    for (int e = wid; e < nEdges; e += nw) {
        const int col = ecol[e]; const int row = erow[e]; const float val = eval[e];
        const float4* src = (const float4*)(support + (size_t)col * D);
        float4 a = src[lane], b = src[lane + 32];
        float* dst = out + (size_t)row * D;
        int o = lane * 4;
        atomicAdd(dst + o + 0, a.x * val); atomicAdd(dst + o + 1, a.y * val);
        atomicAdd(dst + o + 2, a.z * val); atomicAdd(dst + o + 3, a.w * val);
        atomicAdd(dst + 128 + o + 0, b.x * val); atomicAdd(dst + 128 + o + 1, b.y * val);
        atomicAdd(dst + 128 + o + 2, b.z * val); atomicAdd(dst + 128 + o + 3, b.w * val);
    }
}

extern "C" void kernel_launch(void* const* d_in, const int* in_sizes, int n_in,
                              void* d_out, int out_size, void* d_ws, size_t ws_size,
                              hipStream_t stream)
{
    const float* X    = (const float*)d_in[0];
    const int*   erow = (const int*)  d_in[1];
    const int*   ecol = (const int*)  d_in[2];
    const float* eval = (const float*)d_in[3];
    const float* W    = (const float*)d_in[4];
    const float* bias = (const float*)d_in[5];

    const int N = in_sizes[0] / D;   // 100000
    const int E = in_sizes[1];       // 3200000
    float* out = (float*)d_out;

    // ---- workspace partition (256B aligned) ----
    size_t off = 0;
    auto carve = [&](size_t bytes) -> char* {
        char* p = (char*)d_ws + off;
        off += (bytes + 255) & ~(size_t)255;
        return p;
    };
    float* S      = (float*)carve((size_t)N * D * sizeof(float));
    int*   counts = (int*)  carve((size_t)N * sizeof(int));
    int*   starts = (int*)  carve((size_t)(N + 1) * sizeof(int));
    int*   cursor = (int*)  carve((size_t)N * sizeof(int));
    int*   bcol   = (int*)  carve((size_t)E * sizeof(int));
    float* bval   = (float*)carve((size_t)E * sizeof(float));
    const bool binned = (off <= ws_size);

    // 1) support = X @ W  (WMMA f32, async-LDS staged)
    dim3 g1((N + 15) / 16, D / 128);
    gcn_gemm_wmma_f32<<<g1, 256, 0, stream>>>(X, W, S, N);

    if (binned) {
        // 2) CSR build: histogram -> scan -> bin
        gcn_zero_i32<<<(N + 255) / 256, 256, 0, stream>>>(counts, N);
        gcn_edge_histogram<<<(E + 255) / 256, 256, 0, stream>>>(erow, counts, E);
        gcn_scan_starts<<<1, 1024, 0, stream>>>(counts, starts, cursor, N);
        gcn_bin_edges<<<(E + 255) / 256, 256, 0, stream>>>(erow, ecol, eval,
                                                           cursor, bcol, bval, E);
        // 3) wave-per-row gather + bias (no f32 atomics)
        int nWaves = N;                                // one wave per row
        gcn_row_gather<<<(nWaves * 32 + 255) / 256, 256, 0, stream>>>(
            S, starts, bcol, bval, bias, out, N);
    } else {
        // Fallback: bias init + f32 atomic scatter
        int nVec4 = N * (D / 4);
        gcn_init_bias<<<(nVec4 + 255) / 256, 256, 0, stream>>>(
            (float4*)out, (const float4*)bias, nVec4);
        gcn_edge_scatter_atomic<<<4096, 256, 0, stream>>>(S, erow, ecol, eval, out, E);
    }
}